// Model_40630390620470
// MI455X (gfx1250) — compile-verified
//
#include <hip/hip_runtime.h>

typedef __attribute__((ext_vector_type(2))) float v2f;
typedef __attribute__((ext_vector_type(8))) float v8f;
typedef __attribute__((ext_vector_type(4))) int   v4i;

#define HAVE_ASYNC_LDS (__has_builtin(__builtin_amdgcn_global_load_async_to_lds_b128) && \
                        __has_builtin(__builtin_amdgcn_s_wait_asynccnt))

namespace {
constexpr int Lseq = 4096;   // sequence length
constexpr int Hh   = 32;     // heads
constexpr int Pp   = 64;     // head dim (M)
constexpr int Nn   = 64;     // state dim (N)
constexpr int CL   = 64;     // K-chunk staged in LDS
constexpr int NC   = Lseq / CL;
constexpr int LROW = 72;     // padded LDS row stride (floats); 2*72 % 64 == 16 -> conflict-free frag reads
}

#if HAVE_ASYNC_LDS
__device__ __forceinline__ void async_copy16(const float* g, float* l) {
  __builtin_amdgcn_global_load_async_to_lds_b128(
      (__attribute__((address_space(1))) v4i*)g,
      (__attribute__((address_space(3))) v4i*)l,
      0, 0);
}
#endif

// out[b,h,p,n] = sum_t exp(Atot - cumsum(A)_t) * X[b,t,h,p] * B[b,t,h,n]
__global__ __launch_bounds__(256) void mamba2_final_state(
    const float* __restrict__ Xg, const float* __restrict__ Ag,
    const float* __restrict__ Bg, float* __restrict__ Og)
{
  __shared__ __align__(16) float wbuf[Lseq];        // decay weights; reused as reduction buffer at end
  __shared__ float scanb[256];
  __shared__ float stot;
  __shared__ __align__(16) float xs[2][CL * LROW];  // double-buffered X tile (weighted at frag time)
  __shared__ __align__(16) float bs[2][CL * LROW];  // double-buffered B tile

  const int tid = threadIdx.x;
  const int bh  = blockIdx.x;
  const int bi  = bh >> 5;
  const int hi  = bh & 31;

  // ---------------- Phase 1: per-(b,h) cumsum of A -> w_t = exp(Atot - G_t) ----------------
  const float* Ap = Ag + (size_t)bi * Lseq * Hh + hi;   // stride Hh between timesteps
  float av[16];
  float part = 0.f;
#pragma unroll
  for (int i = 0; i < 16; ++i) { av[i] = Ap[(size_t)(tid * 16 + i) * Hh]; part += av[i]; }
  scanb[tid] = part;
  __syncthreads();
  if (tid == 0) {
    float run = 0.f;
#pragma clang loop unroll(disable)
    for (int i = 0; i < 256; ++i) { float v = scanb[i]; scanb[i] = run; run += v; }
    stot = run;
  }
  __syncthreads();
  {
    float run = scanb[tid];
    float tot = stot;
#pragma unroll
    for (int i = 0; i < 16; ++i) {
      run += av[i];
      wbuf[tid * 16 + i] = expf(tot - run);
    }
  }
  // visibility of wbuf is covered by the first barrier inside the chunk loop

  // ---------------- Phase 2: K-chunked WMMA GEMM ----------------
  const int lane = tid & 31;
  const int wid  = tid >> 5;
  const int lm   = lane & 15;         // M/N index within a 16-wide tile
  const int kh   = lane >> 4;         // K-half select (frag holds K = k0+2*kh, k0+2*kh+1)
  const int kg   = wid >> 2;          // K-group: waves 0-3 -> t[0,32), waves 4-7 -> t[32,64)
  const int wq   = wid & 3;
  const int pbase = (wq >> 1) * 32;   // 32x32 output quadrant
  const int nbase = (wq & 1) * 32;

  const size_t tstr = (size_t)Hh * Pp;                      // 2048 floats between timesteps
  const float* Xb = Xg + ((size_t)bi * Lseq * Hh + hi) * Pp;
  const float* Bb = Bg + ((size_t)bi * Lseq * Hh + hi) * Nn;

  v8f acc00 = {0,0,0,0,0,0,0,0};
  v8f acc01 = {0,0,0,0,0,0,0,0};
  v8f acc10 = {0,0,0,0,0,0,0,0};
  v8f acc11 = {0,0,0,0,0,0,0,0};

  // per-thread staging slice: 4 float4 of X + 4 of B per chunk (32 KB/chunk total)
  const int row0 = tid >> 4;                 // rows row0 + 16*k
  const int c4   = (tid & 15) << 2;          // float offset within a row

#if HAVE_ASYNC_LDS
  // ---- CDNA5 async global->LDS staging (ASYNCcnt-tracked DMA) ----
  {
    // prologue: issue chunk 0
    float* xsp = xs[0];
    float* bsp = bs[0];
#pragma unroll
    for (int k = 0; k < 4; ++k) {
      const int row = row0 + (k << 4);
      async_copy16(Xb + (size_t)row * tstr + c4, xsp + row * LROW + c4);
      async_copy16(Bb + (size_t)row * tstr + c4, bsp + row * LROW + c4);
    }
  }
  for (int c = 0; c < NC; ++c) {
    float* xsp = xs[c & 1];
    float* bsp = bs[c & 1];
    __builtin_amdgcn_s_wait_asynccnt(0);   // my chunk-c transfers are in LDS
    __syncthreads();                        // everyone's transfers visible; buf[1-cur] free
    if (c + 1 < NC) {
      const float* xn = Xb + (size_t)(c + 1) * CL * tstr;
      const float* bn = Bb + (size_t)(c + 1) * CL * tstr;
      float* xd = xs[(c + 1) & 1];
      float* bd = bs[(c + 1) & 1];
#pragma unroll
      for (int k = 0; k < 4; ++k) {
        const int row = row0 + (k << 4);
        async_copy16(xn + (size_t)row * tstr + c4, xd + row * LROW + c4);
        async_copy16(bn + (size_t)row * tstr + c4, bd + row * LROW + c4);
      }
    }
#else
  // ---- fallback: software-pipelined register staging ----
  float4 rx[4], rb4[4];
#pragma unroll
  for (int k = 0; k < 4; ++k) {
    const int row = row0 + (k << 4);
    rx[k]  = *(const float4*)(Xb + (size_t)row * tstr + c4);
    rb4[k] = *(const float4*)(Bb + (size_t)row * tstr + c4);
  }
  for (int c = 0; c < NC; ++c) {
    float* xsp = xs[c & 1];
    float* bsp = bs[c & 1];
#pragma unroll
    for (int k = 0; k < 4; ++k) {
      const int row = row0 + (k << 4);
      *(float4*)(xsp + row * LROW + c4) = rx[k];
      *(float4*)(bsp + row * LROW + c4) = rb4[k];
    }
    if (c + 1 < NC) {
      const float* xn = Xb + (size_t)(c + 1) * CL * tstr;
      const float* bn = Bb + (size_t)(c + 1) * CL * tstr;
#pragma unroll
      for (int k = 0; k < 4; ++k) {
        const int row = row0 + (k << 4);
        rx[k]  = *(const float4*)(xn + (size_t)row * tstr + c4);
        rb4[k] = *(const float4*)(bn + (size_t)row * tstr + c4);
      }
    }
    __syncthreads();   // buf[c&1] ready for everyone; buf[1-(c&1)] free
#endif

    // ---- compute on buf[c&1] ----
    const float* wp = wbuf + c * CL;
    const int kt0 = kg * 32;
#pragma unroll
    for (int ks = 0; ks < 8; ++ks) {
      const int k0 = kt0 + (ks << 2) + (kh << 1);   // local t for this lane's K pair
      const float wx = wp[k0];
      const float wy = wp[k0 + 1];
      const float* xr0 = xsp + k0 * LROW;
      const float* xr1 = xsp + (k0 + 1) * LROW;
      const float* br0 = bsp + k0 * LROW;
      const float* br1 = bsp + (k0 + 1) * LROW;
      v2f a0, a1, f0, f1;
      a0.x = xr0[pbase + lm] * wx;        a0.y = xr1[pbase + lm] * wy;
      a1.x = xr0[pbase + 16 + lm] * wx;   a1.y = xr1[pbase + 16 + lm] * wy;
      f0.x = br0[nbase + lm];             f0.y = br1[nbase + lm];
      f1.x = br0[nbase + 16 + lm];        f1.y = br1[nbase + 16 + lm];
      acc00 = __builtin_amdgcn_wmma_f32_16x16x4_f32(false, a0, false, f0, (short)0, acc00, false, false);
      acc01 = __builtin_amdgcn_wmma_f32_16x16x4_f32(false, a0, false, f1, (short)0, acc01, false, false);
      acc10 = __builtin_amdgcn_wmma_f32_16x16x4_f32(false, a1, false, f0, (short)0, acc10, false, false);
      acc11 = __builtin_amdgcn_wmma_f32_16x16x4_f32(false, a1, false, f1, (short)0, acc11, false, false);
    }
  }

  // ---------------- Phase 3: reduce the two K-groups, store ----------------
  __syncthreads();                                   // all LDS / wbuf reads done; reuse wbuf (16 KB = 16 tiles x 256)
  float* red = wbuf;
  const int tadd = (pbase >> 4) * 4 + (nbase >> 4);  // base 16x16 tile index (4x4 tile grid)
  if (kg == 1) {
#pragma unroll
    for (int r = 0; r < 8; ++r) {
      red[(tadd + 0) * 256 + lane * 8 + r] = acc00[r];
      red[(tadd + 1) * 256 + lane * 8 + r] = acc01[r];
      red[(tadd + 4) * 256 + lane * 8 + r] = acc10[r];
      red[(tadd + 5) * 256 + lane * 8 + r] = acc11[r];
    }
  }
  __syncthreads();
  if (kg == 0) {
    float* op = Og + (size_t)bh * Pp * Nn;           // out[b,h,:,:], n contiguous
#pragma unroll
    for (int r = 0; r < 8; ++r) {
      const int m = r + kh * 8;                      // C/D layout: VGPR r -> M=r (+8 for upper lanes)
      op[(pbase + m) * Nn + nbase + lm]           = acc00[r] + red[(tadd + 0) * 256 + lane * 8 + r];
      op[(pbase + m) * Nn + nbase + 16 + lm]      = acc01[r] + red[(tadd + 1) * 256 + lane * 8 + r];
      op[(pbase + 16 + m) * Nn + nbase + lm]      = acc10[r] + red[(tadd + 4) * 256 + lane * 8 + r];
      op[(pbase + 16 + m) * Nn + nbase + 16 + lm] = acc11[r] + red[(tadd + 5) * 256 + lane * 8 + r];
    }
  }
}

extern "C" void kernel_launch(void* const* d_in, const int* in_sizes, int n_in,
                              void* d_out, int out_size, void* d_ws, size_t ws_size,
                              hipStream_t stream) {
  (void)in_sizes; (void)n_in; (void)d_ws; (void)ws_size; (void)out_size;
  const float* X = (const float*)d_in[0];
  const float* A = (const float*)d_in[1];
  const float* B = (const float*)d_in[2];
  // d_in[3] (C) is unused by the reference output; d_in[4] (block_len) does not change the math.
  float* out = (float*)d_out;
  mamba2_final_state<<<dim3(8 * 32), dim3(256), 0, stream>>>(X, A, B, out);
}